// CapsuleRouting_8667244003371
// MI455X (gfx1250) — compile-verified
//
#include <hip/hip_runtime.h>
#include <cstdint>

// Problem constants (from reference): b=16, B=288, C=32, P2=16, f*f=36
#define NBATCH 16
#define NB_IN  288
#define NC     32
#define NP2    16
#define NF2    36
#define CELL   576   // NP2*NF2 floats per "cellblock" (one (b,B,C) tile), 2304 bytes

typedef __attribute__((ext_vector_type(2))) float v2f;
typedef __attribute__((ext_vector_type(8))) float v8f;

// ---- CDNA5 async global->LDS copy (gfx1250), tracked by ASYNCcnt ------------
// Per-lane: copies 16 bytes from the global address saddr plus vaddr32 to the LDS address in vdst32.
__device__ __forceinline__ void async_b128(const void* lds_dst, const void* gbase,
                                           uint32_t voff) {
  uint32_t l = (uint32_t)(uintptr_t)lds_dst;      // low 32 bits of generic LDS ptr are the LDS offset
  uint64_t s = (uint64_t)(uintptr_t)gbase;        // uniform base -> SGPR pair
  asm volatile("global_load_async_to_lds_b128 %0, %1, %2"
               :: "v"(l), "v"(voff), "s"(s) : "memory");
}
__device__ __forceinline__ void async_wait0() {
  asm volatile("s_wait_asynccnt 0x0" ::: "memory");
}

// ---- kernel 1: sm[b,B,fc] = softmax over B of a[b,B,fc] ---------------------
__global__ void k_softmax_a(const float* __restrict__ a, float* __restrict__ sm) {
  int blk = blockIdx.x;                 // 0..575 = b*36+fc
  int b = blk / NF2, fc = blk % NF2;
  int t = threadIdx.x;                  // 0..287
  __shared__ float red[NB_IN];
  __shared__ float red2[32];
  size_t idx = ((size_t)(b * NB_IN + t)) * NF2 + fc;
  float x = a[idx];
  red[t] = x; __syncthreads();
  if (t < 32) { float m = -1e30f; for (int i = t; i < NB_IN; i += 32) m = fmaxf(m, red[i]); red2[t] = m; }
  __syncthreads();
  if (t == 0) { float m = -1e30f; for (int i = 0; i < 32; i++) m = fmaxf(m, red2[i]); red2[0] = m; }
  __syncthreads();
  float mx = red2[0];
  float e = expf(x - mx);
  __syncthreads();
  red[t] = e; __syncthreads();
  if (t < 32) { float ss = 0.f; for (int i = t; i < NB_IN; i += 32) ss += red[i]; red2[t] = ss; }
  __syncthreads();
  if (t == 0) { float ss = 0.f; for (int i = 0; i < 32; i++) ss += red2[i]; red2[0] = ss; }
  __syncthreads();
  sm[idx] = e / red2[0];
}

// ---- kernel 2: s_raw[b,C,P2,fc] = sum_B c[b,B,(C),fc] * u_raw ---------------
// Block per (b,C); 576 threads, thread t owns output (p=t/36, fc=t%36).
// Streams u via async double-buffered LDS (4 B-cellblocks / round).
// use_sm=1 (iter 0): coupling = sm[b,B,fc] and additionally computes
//   scale[b,C,fc] = 1/(max_B ||u||_P2 - min_B ||u||_P2)  (fused norm pass).
__global__ void __launch_bounds__(CELL) k_accum_s(
    const float* __restrict__ u, const float* __restrict__ csrc,
    float* __restrict__ s_out, float* __restrict__ scale_out, int use_sm) {
  __shared__ float u_buf[2][4 * CELL];   // 2 x 9216 B
  __shared__ float c_buf[2][4 * NF2];
  __shared__ float redmx[144], redmn[144];
  int bc = blockIdx.x;                  // b*32+C
  int b = bc / NC, C = bc % NC;
  int t = threadIdx.x;                  // 0..575
  const float* ubase = u + ((size_t)b * NB_IN * NC + C) * CELL;
  const float* cbase = use_sm ? (csrc + (size_t)b * NB_IN * NF2)
                              : (csrc + ((size_t)b * NB_IN * NC + C) * NF2);
  int cBstride = use_sm ? NF2 : (NC * NF2);  // floats between B rows of coupling
  int bl_u = t / 144, seg_u = t % 144;
  // round 0 issue
  async_b128(&u_buf[0][t * 4], ubase, (uint32_t)(bl_u * NC * CELL * 4 + seg_u * 16));
  if (t < 36) {
    int bl = t / 9, seg = t % 9;
    async_b128(&c_buf[0][t * 4], cbase, (uint32_t)(bl * cBstride * 4 + seg * 16));
  }
  async_wait0(); __syncthreads();

  float acc = 0.f, mx = -1e30f, mn = 1e30f;
  int fc = t % NF2;
  const int R = NB_IN / 4;              // 72 rounds
  for (int r = 0; r < R; ++r) {
    int cb = r & 1, nb = cb ^ 1;
    if (r + 1 < R) {
      async_b128(&u_buf[nb][t * 4], ubase,
                 (uint32_t)(((r + 1) * 4 + bl_u) * NC * CELL * 4 + seg_u * 16));
      if (t < 36) {
        int bl = t / 9, seg = t % 9;
        async_b128(&c_buf[nb][t * 4], cbase,
                   (uint32_t)((((r + 1) * 4 + bl) * cBstride) * 4 + seg * 16));
      }
    }
#pragma unroll
    for (int bl = 0; bl < 4; ++bl)
      acc += c_buf[cb][bl * NF2 + fc] * u_buf[cb][bl * CELL + t];
    if (use_sm && t < 144) {            // fused max-min-norm statistics
      int bln = t / NF2;
      float n2 = 0.f;
#pragma unroll
      for (int p = 0; p < NP2; ++p) {
        float x = u_buf[cb][bln * CELL + p * NF2 + fc];
        n2 += x * x;
      }
      float nrm = sqrtf(n2);
      mx = fmaxf(mx, nrm); mn = fminf(mn, nrm);
    }
    if (r + 1 < R) async_wait0();
    __syncthreads();
  }
  s_out[(size_t)bc * CELL + t] = acc;
  if (use_sm) {
    if (t < 144) { redmx[t] = mx; redmn[t] = mn; }
    __syncthreads();
    if (t < NF2) {
      float M = fmaxf(fmaxf(redmx[t], redmx[t + 36]), fmaxf(redmx[t + 72], redmx[t + 108]));
      float m = fminf(fminf(redmn[t], redmn[t + 36]), fminf(redmn[t + 72], redmn[t + 108]));
      scale_out[(size_t)bc * NF2 + t] = 1.0f / (M - m);
    }
  }
}

// ---- kernel 3: squash + scale; emits vs = scale*v (or final v, a_out) -------
// One wave handles 16 pose-columns; sum_P2 of sc*s (not squared) via v_wmma_f32_16x16x4_f32 with
// an all-ones A matrix (D[.,N] = sum_K B[K,N], invariant to K permutation).
__global__ void k_squash(const float* __restrict__ s, const float* __restrict__ scale,
                         float* __restrict__ vs, float* __restrict__ out, int last) {
  int tid = threadIdx.x;
  int wave = tid >> 5, lane = tid & 31;
  int tile = blockIdx.x * 8 + wave;     // 0..1151
  int col = tile * 16 + (lane & 15);    // 0..18431 ; col = slice*36 + fc
  int slice = col / NF2, fc = col % NF2;
  size_t base = (size_t)slice * CELL + fc;
  float sc = scale[col];
  int hi = (lane >= 16) ? 1 : 0;
  float sv[8];
#pragma unroll
  for (int q = 0; q < 4; q++) {
    sv[2 * q]     = sc * s[base + (size_t)(4 * q + (hi ? 2 : 0)) * NF2];
    sv[2 * q + 1] = sc * s[base + (size_t)(4 * q + (hi ? 3 : 1)) * NF2];
  }
  v2f ones; ones[0] = 1.0f; ones[1] = 1.0f;
  v8f acc = {0.f, 0.f, 0.f, 0.f, 0.f, 0.f, 0.f, 0.f};
#pragma unroll
  for (int q = 0; q < 4; q++) {
    v2f bm; bm[0] = sv[2 * q]; bm[1] = sv[2 * q + 1];
    acc = __builtin_amdgcn_wmma_f32_16x16x4_f32(false, ones, false, bm,
                                                (short)0, acc, false, false);
  }
  float sn = acc[0];                     // sum_P2 s_true (not squared) for this column
  float factor = sn / ((1.0f + sn) * sqrtf(sn + 1e-5f));
#pragma unroll
  for (int q = 0; q < 4; q++) {
    int p0 = 4 * q + (hi ? 2 : 0), p1 = 4 * q + (hi ? 3 : 1);
    float v0 = factor * sv[2 * q], v1 = factor * sv[2 * q + 1];
    if (!last) {
      vs[base + (size_t)p0 * NF2] = sc * v0;
      vs[base + (size_t)p1 * NF2] = sc * v1;
    } else {
      out[base + (size_t)p0 * NF2] = v0;
      out[base + (size_t)p1 * NF2] = v1;
    }
  }
  if (last && lane < 16)
    out[(size_t)NBATCH * NC * CELL + col] = factor * sqrtf(sn);
}

// ---- kernel 4: r += sum_P2 u*vs ; then c = softmax_C(r)*sm*C (fused) --------
// Block per (b,B); streams 32 contiguous C-cellblocks via async double buffer,
// preloads vs[b] slice + r row + sm row via async copies too.
__global__ void __launch_bounds__(CELL) k_update_r(
    const float* __restrict__ u, const float* __restrict__ vsb,
    const float* __restrict__ sm, float* __restrict__ rg, float* __restrict__ cg,
    int first) {
  __shared__ float vs_l[NC * CELL];      // 73728 B
  __shared__ float u_buf[2][4 * CELL];   // 18432 B
  __shared__ float r_l[NC * NF2];        // 4608 B
  __shared__ float sm_l[NF2];
  __shared__ float mx_l[NF2], inv_l[NF2];
  int blk = blockIdx.x;
  int b = blk / NB_IN, B = blk % NB_IN;
  int t = threadIdx.x;                   // 0..575
  const float* ubase = u + ((size_t)b * NB_IN + B) * NC * CELL;
  const float* vsbase = vsb + (size_t)b * NC * CELL;
  const float* rbase = rg + ((size_t)b * NB_IN + B) * NC * NF2;
  const float* smbase = sm + ((size_t)b * NB_IN + B) * NF2;
#pragma unroll
  for (int i = 0; i < 8; i++)
    async_b128(&vs_l[(i * CELL + t) * 4], vsbase, (uint32_t)((i * CELL + t) * 16));
  if (!first && t < 288) async_b128(&r_l[t * 4], rbase, (uint32_t)(t * 16));
  if (t < 9) async_b128(&sm_l[t * 4], smbase, (uint32_t)(t * 16));
  async_b128(&u_buf[0][t * 4], ubase, (uint32_t)(t * 16));  // round 0 (C=0..3)
  async_wait0(); __syncthreads();
  if (first && t < 288) {
#pragma unroll
    for (int k = 0; k < 4; k++) r_l[t * 4 + k] = 0.f;
  }
  __syncthreads();

  const int R2 = NC / 4;                 // 8 rounds
  for (int r = 0; r < R2; ++r) {
    int cb = r & 1, nb = cb ^ 1;
    if (r + 1 < R2)
      async_b128(&u_buf[nb][t * 4], ubase, (uint32_t)((r + 1) * 9216 + t * 16));
    if (t < 144) {
      int cl = t / NF2, fc = t % NF2;
      int Ci = r * 4 + cl;
      float dot = 0.f;
#pragma unroll
      for (int p = 0; p < NP2; ++p)
        dot += u_buf[cb][cl * CELL + p * NF2 + fc] * vs_l[Ci * CELL + p * NF2 + fc];
      float rn = r_l[Ci * NF2 + fc] + dot;     // vs already carries scale
      r_l[Ci * NF2 + fc] = rn;
      rg[((size_t)b * NB_IN + B) * NC * NF2 + Ci * NF2 + fc] = rn;
    }
    if (r + 1 < R2) async_wait0();
    __syncthreads();
  }
  // softmax over C (fc-wise), then write c = softmax * sm * C
  if (t < NF2) {
    float m = -1e30f;
    for (int Ci = 0; Ci < NC; Ci++) m = fmaxf(m, r_l[Ci * NF2 + t]);
    float ss = 0.f;
    for (int Ci = 0; Ci < NC; Ci++) ss += expf(r_l[Ci * NF2 + t] - m);
    mx_l[t] = m; inv_l[t] = 1.0f / ss;
  }
  __syncthreads();
  size_t cbase = ((size_t)b * NB_IN + B) * NC * NF2;
#pragma unroll
  for (int k = 0; k < 2; k++) {
    int idx = t + k * CELL;              // 0..1151
    int fc = idx % NF2;
    cg[cbase + idx] = expf(r_l[idx] - mx_l[fc]) * inv_l[fc] * sm_l[fc] * (float)NC;
  }
}

// ---- host orchestration -----------------------------------------------------
extern "C" void kernel_launch(void* const* d_in, const int* in_sizes, int n_in,
                              void* d_out, int out_size, void* d_ws, size_t ws_size,
                              hipStream_t stream) {
  (void)in_sizes; (void)n_in; (void)out_size; (void)ws_size;
  const float* u = (const float*)d_in[0];
  const float* a = (const float*)d_in[1];
  float* out = (float*)d_out;
  float* ws = (float*)d_ws;
  float* sm    = ws;                 // 165888
  float* scale = ws + 165888;        // 18432
  float* sbuf  = ws + 184320;        // 294912
  float* vs    = ws + 479232;        // 294912
  float* r     = ws + 774144;        // 5308416
  float* c     = ws + 6082560;       // 5308416 (total 45.6 MB)

  k_softmax_a<<<576, 288, 0, stream>>>(a, sm);
  // iter 0 (c = softmax(a); fused norm/scale pass)
  k_accum_s<<<512, CELL, 0, stream>>>(u, sm, sbuf, scale, 1);
  k_squash<<<144, 256, 0, stream>>>(sbuf, scale, vs, out, 0);
  k_update_r<<<NBATCH * NB_IN, CELL, 0, stream>>>(u, vs, sm, r, c, 1);
  // iter 1
  k_accum_s<<<512, CELL, 0, stream>>>(u, c, sbuf, scale, 0);
  k_squash<<<144, 256, 0, stream>>>(sbuf, scale, vs, out, 0);
  k_update_r<<<NBATCH * NB_IN, CELL, 0, stream>>>(u, vs, sm, r, c, 0);
  // iter 2 (final)
  k_accum_s<<<512, CELL, 0, stream>>>(u, c, sbuf, scale, 0);
  k_squash<<<144, 256, 0, stream>>>(sbuf, scale, vs, out, 1);
}